// LGNGuard_70300024701476
// MI455X (gfx1250) — compile-verified
//
#include <hip/hip_runtime.h>
#include <math.h>

#define N_USER 60000
#define N_ITEM 40000
#define NN     100000
#define DIM    64
#define EDGES  800000
#define BATCH  2048
#define EPSV   1e-8f
#define BETA   0.5f

typedef __attribute__((ext_vector_type(2))) float v2f;
typedef __attribute__((ext_vector_type(8))) float v8f;

// ---- init: emb = concat(user_emb, item_emb); acc = emb ----
__global__ void k_init_emb(const float* __restrict__ ue, const float* __restrict__ ie,
                           float* __restrict__ emb, float* __restrict__ acc) {
    int i = blockIdx.x * 256 + threadIdx.x;
    if (i < NN * DIM) {
        float v = (i < N_USER * DIM) ? ue[i] : ie[i - N_USER * DIM];
        emb[i] = v;
        acc[i] = v;
    }
}

// ---- per-edge cosine similarity, 16 lanes per edge (2 edges per wave32) ----
// also accumulates rowsum[src] = sum |cos| via f32 atomics (rowsum pre-zeroed)
__global__ void k_cos(const float* __restrict__ emb, const int* __restrict__ src,
                      const int* __restrict__ dst, float* __restrict__ cosv,
                      float* __restrict__ rowsum) {
    int tid = blockIdx.x * 256 + threadIdx.x;
    int e = tid >> 4;
    if (e >= EDGES) return;
    int l = threadIdx.x & 15;
    int s = src[e], d0 = dst[e];
    float4 a = *(const float4*)(emb + (size_t)s  * DIM + l * 4);
    float4 b = *(const float4*)(emb + (size_t)d0 * DIM + l * 4);
    float dot = a.x * b.x + a.y * b.y + a.z * b.z + a.w * b.w;
    float na  = a.x * a.x + a.y * a.y + a.z * a.z + a.w * a.w;
    float nb  = b.x * b.x + b.y * b.y + b.z * b.z + b.w * b.w;
    #pragma unroll
    for (int off = 8; off >= 1; off >>= 1) {
        dot += __shfl_xor(dot, off, 16);
        na  += __shfl_xor(na,  off, 16);
        nb  += __shfl_xor(nb,  off, 16);
    }
    if (l == 0) {
        float den = fmaxf(sqrtf(na) * sqrtf(nb), EPSV);
        float c = dot / den;
        cosv[e] = c;
        atomicAdd(rowsum + s, fabsf(c));
    }
}

// ---- row-wise L1 normalize (in place on cosv) ----
__global__ void k_coef(const int* __restrict__ src, float* __restrict__ cosv,
                       const float* __restrict__ rowsum) {
    int e = blockIdx.x * 256 + threadIdx.x;
    if (e >= EDGES) return;
    float rs = rowsum[src[e]];
    cosv[e] = cosv[e] / (rs > 0.0f ? rs : 1.0f);
}

// ---- prune (sigmoid(Linear([c, c_rev])) > 0.5  <=>  z > 0), EMA memory, g = mem*adj ----
__global__ void k_prune(const float* __restrict__ cosv, const int* __restrict__ rev,
                        const float* __restrict__ W, const float* __restrict__ bp,
                        const float* __restrict__ adj, float* __restrict__ mem,
                        float* __restrict__ g) {
    int e = blockIdx.x * 256 + threadIdx.x;
    if (e >= EDGES) return;
    float c  = cosv[e];
    float cr = cosv[rev[e]];
    float z  = c * W[0] + cr * W[1] + bp[0];
    float cc = (z > 0.0f) ? c : 0.0f;          // == sigmoid(z) > 0.5
    float m  = BETA * mem[e] + (1.0f - BETA) * cc;
    mem[e] = m;
    g[e]   = m * adj[e];
}

// ---- SpMM scatter: out[src[e], :] += g[e] * emb[dst[e], :]  (64 lanes per edge) ----
__global__ void k_spmm(const float* __restrict__ emb, const float* __restrict__ g,
                       const int* __restrict__ src, const int* __restrict__ dst,
                       float* __restrict__ out) {
    int tid = blockIdx.x * 256 + threadIdx.x;
    int e = tid >> 6;
    if (e >= EDGES) return;
    int d = tid & 63;
    float val = g[e] * emb[(size_t)dst[e] * DIM + d];
    atomicAdd(out + (size_t)src[e] * DIM + d, val);
}

// ---- acc += layer output ----
__global__ void k_acc(float* __restrict__ acc, const float* __restrict__ nxt) {
    int i = blockIdx.x * 256 + threadIdx.x;
    if (i < NN * DIM) acc[i] += nxt[i];
}

// ---- gather batch rows with mean scaling (mean over 4 embeddings = *0.25) ----
__global__ void k_gather(const float* __restrict__ acc, const int* __restrict__ idx,
                         int base, float* __restrict__ out) {
    int i = blockIdx.x * 256 + threadIdx.x;
    if (i >= BATCH * DIM) return;
    int b = i >> 6, d = i & 63;
    out[i] = 0.25f * acc[(size_t)(idx[b] + base) * DIM + d];
}

// ---- final GEMM + sigmoid via V_WMMA_F32_16X16X4_F32, one wave per 16x16 tile ----
// A: U tile (16xK), B: I-tile transposed (Kx16) with B[k][n] = I[n][k].
__global__ __launch_bounds__(32) void k_gemm_sig(const float* __restrict__ U,
                                                 const float* __restrict__ I,
                                                 float* __restrict__ out) {
    const int lane = threadIdx.x;        // 0..31
    const int half = lane >> 4;          // 0 or 1
    const int l16  = lane & 15;
    const float* urow = U + (size_t)(blockIdx.x * 16 + l16) * DIM;  // A row M = l16
    const float* irow = I + (size_t)(blockIdx.y * 16 + l16) * DIM;  // B col N = l16
    v8f c = {};
    #pragma unroll
    for (int k0 = 0; k0 < DIM; k0 += 4) {
        int kk = k0 + 2 * half;          // v0: K=k0 / k0+2; v1: K=k0+1 / k0+3
        float2 av = *(const float2*)(urow + kk);
        float2 bv = *(const float2*)(irow + kk);
        v2f a; a.x = av.x; a.y = av.y;
        v2f b; b.x = bv.x; b.y = bv.y;
        c = __builtin_amdgcn_wmma_f32_16x16x4_f32(false, a, false, b,
                                                  (short)0, c, false, false);
    }
    // C/D layout: VGPR r -> M = r + 8*half, N = l16
    #pragma unroll
    for (int r = 0; r < 8; ++r) {
        int row = blockIdx.x * 16 + r + 8 * half;
        int col = blockIdx.y * 16 + l16;
        float v = c[r];
        out[(size_t)row * BATCH + col] = 1.0f / (1.0f + __expf(-v));
    }
}

extern "C" void kernel_launch(void* const* d_in, const int* in_sizes, int n_in,
                              void* d_out, int out_size, void* d_ws, size_t ws_size,
                              hipStream_t stream) {
    const int*   users    = (const int*)d_in[0];
    const int*   items    = (const int*)d_in[1];
    const float* user_emb = (const float*)d_in[2];
    const float* item_emb = (const float*)d_in[3];
    const float* W_prune  = (const float*)d_in[4];
    const float* b_prune  = (const float*)d_in[5];
    const int*   src      = (const int*)d_in[6];
    const int*   dst      = (const int*)d_in[7];
    const int*   rev      = (const int*)d_in[8];
    const float* adj      = (const float*)d_in[9];
    float* out = (float*)d_out;

    float* p = (float*)d_ws;
    float* embA   = p; p += (size_t)NN * DIM;
    float* embB   = p; p += (size_t)NN * DIM;
    float* acc    = p; p += (size_t)NN * DIM;
    float* mem    = p; p += EDGES;
    float* cosv   = p; p += EDGES;
    float* gbuf   = p; p += EDGES;
    float* rowsum = p; p += NN;
    float* Ub     = p; p += BATCH * DIM;
    float* Ib     = p; p += BATCH * DIM;

    k_init_emb<<<(NN * DIM + 255) / 256, 256, 0, stream>>>(user_emb, item_emb, embA, acc);
    hipMemcpyAsync(mem, adj, EDGES * sizeof(float), hipMemcpyDeviceToDevice, stream);

    float* cur = embA;
    float* nxt = embB;
    for (int layer = 0; layer < 3; ++layer) {
        hipMemsetAsync(rowsum, 0, NN * sizeof(float), stream);
        hipMemsetAsync(nxt, 0, (size_t)NN * DIM * sizeof(float), stream);
        k_cos  <<<(EDGES * 16 + 255) / 256, 256, 0, stream>>>(cur, src, dst, cosv, rowsum);
        k_coef <<<(EDGES + 255) / 256, 256, 0, stream>>>(src, cosv, rowsum);
        k_prune<<<(EDGES + 255) / 256, 256, 0, stream>>>(cosv, rev, W_prune, b_prune, adj, mem, gbuf);
        k_spmm <<<(EDGES * 64 + 255) / 256, 256, 0, stream>>>(cur, gbuf, src, dst, nxt);
        k_acc  <<<(NN * DIM + 255) / 256, 256, 0, stream>>>(acc, nxt);
        float* t = cur; cur = nxt; nxt = t;
    }

    k_gather<<<(BATCH * DIM + 255) / 256, 256, 0, stream>>>(acc, users, 0, Ub);
    k_gather<<<(BATCH * DIM + 255) / 256, 256, 0, stream>>>(acc, items, N_USER, Ib);

    dim3 g(BATCH / 16, BATCH / 16);
    k_gemm_sig<<<g, 32, 0, stream>>>(Ub, Ib, out);
}